// QNetwork_52037823758431
// MI455X (gfx1250) — compile-verified
//
#include <hip/hip_runtime.h>

typedef __attribute__((ext_vector_type(2))) float v2f;
typedef __attribute__((ext_vector_type(8))) float v8f;

// ---------------------------------------------------------------------------
// Elementwise fill
// ---------------------------------------------------------------------------
__global__ __launch_bounds__(256) void fill_f32(float* __restrict__ p, float v, long n) {
    long i = (long)blockIdx.x * 256 + threadIdx.x;
    if (i < n) p[i] = v;
}

// ---------------------------------------------------------------------------
// Degree count (self-loop handled by init deg=1.0)
// ---------------------------------------------------------------------------
__global__ __launch_bounds__(256) void deg_count(const int* __restrict__ dst,
                                                 float* __restrict__ deg, int nE) {
    int e = blockIdx.x * 256 + threadIdx.x;
    if (e < nE) atomicAdd(&deg[dst[e]], 1.0f);
}

__global__ __launch_bounds__(256) void rsqrt_k(const float* __restrict__ deg,
                                               float* __restrict__ dinv, int n) {
    int i = blockIdx.x * 256 + threadIdx.x;
    if (i < n) dinv[i] = rsqrtf(deg[i]);
}

// ---------------------------------------------------------------------------
// WMMA GEMM: out[M x N] = A[M x 128] @ W[128 x N] (+bias) (optional relu)
// grid.x = ceil(M/16); block = 256 threads = 8 waves; wave w owns col tile(s).
// V_WMMA_F32_16X16X4_F32 (fp32 in / fp32 accumulate) matches fp32 reference.
// Both operands staged in LDS as v2f with row pitch 66 (=132 floats):
//  - each A/B fragment is ONE aligned ds_load_b64
//  - bank = (132*r + kk) % 64 = (4r + kk) % 64: half-wave lo hits {0,4,..,60},
//    half-wave hi (kk+2) hits {2,6,..,62} -> conflict-free b64 reads.
// W staged TRANSPOSED so W[kk][n], W[kk+1][n] are adjacent.
// ---------------------------------------------------------------------------
#define LPITCH 66  // v2f per row = 132 floats

__global__ __launch_bounds__(256) void gemm_k128(const float* __restrict__ A,
                                                 const float* __restrict__ W,
                                                 const float* __restrict__ bias,
                                                 float* __restrict__ out,
                                                 int M, int N, int do_relu, int has_bias) {
    __shared__ v2f sA2[16][LPITCH];    // [row][k/2]
    __shared__ v2f sW2[128][LPITCH];   // [col][k/2]  (transposed W)

    const int row0 = blockIdx.x * 16;
    const int tid  = threadIdx.x;

    // Stage W (128 x N, row-major in memory) transposed into LDS.
    // i = k*N + n -> sW2[n][k/2].{x,y}. Consecutive threads: same k, n+1 ->
    // LDS bank steps by 4 -> conflict-free.
    float* sWf = (float*)sW2;
    for (int i = tid; i < 128 * N; i += 256) {
        int k = i / N, n = i - k * N;
        sWf[n * (2 * LPITCH) + k] = W[i];
    }
    // Stage A strip (16 rows x 128), coalesced reads; clamp OOB rows to zero.
    float* sAf = (float*)sA2;
    for (int i = tid; i < 16 * 128; i += 256) {
        int r = i >> 7, c = i & 127;
        sAf[r * (2 * LPITCH) + c] = (row0 + r < M) ? A[(size_t)(row0 + r) * 128 + c] : 0.0f;
    }
    __syncthreads();

    const int wave  = tid >> 5;
    const int lane  = tid & 31;
    const int m     = lane & 15;        // row (A) / col (B,C,D) within tile
    const int hsel  = lane >> 4;        // 0: K pair {k,k+1}; 1: {k+2,k+3}
    const int mb    = hsel << 3;        // C/D row base: 0 or 8
    const int ncolt = N >> 4;

    for (int ct = wave; ct < ncolt; ct += 8) {
        const int col0 = ct << 4;
        v8f acc = {};
#pragma unroll 4
        for (int k2 = 0; k2 < 64; k2 += 2) {       // k2 = k/2, step = 4 K per WMMA
            const int kq = k2 + hsel;              // v2f index: (k + 2*hsel)/2
            v2f a = sA2[m][kq];                    // one ds_load_b64
            v2f b = sW2[col0 + m][kq];             // one ds_load_b64
            acc = __builtin_amdgcn_wmma_f32_16x16x4_f32(
                false, a, false, b, (short)0, acc, false, false);
        }
        const float bv = has_bias ? bias[col0 + m] : 0.0f;
#pragma unroll
        for (int r = 0; r < 8; ++r) {
            int row = row0 + mb + r;   // C/D layout: VGPR r -> M = r (+8 hi half)
            if (row < M) {
                float v = acc[r] + bv;
                if (do_relu) v = fmaxf(v, 0.0f);
                out[(size_t)row * N + col0 + m] = v;
            }
        }
    }
}

// ---------------------------------------------------------------------------
// Edge scatter: agg[dst] += dinv[src]*dinv[dst] * h[src]
// One wave per edge; each lane handles one float4 (32 lanes x 4 = 128 feats).
// h (51 MB) is L2-resident, so the random gather + atomics live in L2.
// ---------------------------------------------------------------------------
__global__ __launch_bounds__(256) void edge_scatter(const int* __restrict__ src,
                                                    const int* __restrict__ dst,
                                                    const float* __restrict__ dinv,
                                                    const float* __restrict__ h,
                                                    float* __restrict__ agg, int nE) {
    int e = blockIdx.x * 8 + (threadIdx.x >> 5);
    if (e >= nE) return;
    const int lane = threadIdx.x & 31;
    const int s = src[e], d = dst[e];
    const float norm = dinv[s] * dinv[d];
    const float4 v = ((const float4*)(h + (size_t)s * 128))[lane];
    float* ap = agg + (size_t)d * 128 + lane * 4;
    atomicAdd(ap + 0, norm * v.x);
    atomicAdd(ap + 1, norm * v.y);
    atomicAdd(ap + 2, norm * v.z);
    atomicAdd(ap + 3, norm * v.w);
}

// ---------------------------------------------------------------------------
// Finalize GCN layer in place over h: h = relu(agg + dinv^2 * h + bias)
// Thread per float4.
// ---------------------------------------------------------------------------
__global__ __launch_bounds__(256) void finalize_k(const float* __restrict__ agg,
                                                  float* __restrict__ h,
                                                  const float* __restrict__ dinv,
                                                  const float* __restrict__ bias,
                                                  int nNodes) {
    long i = (long)blockIdx.x * 256 + threadIdx.x;
    long total = (long)nNodes * 32;
    if (i >= total) return;
    const int node = (int)(i >> 5);
    const int c    = (int)(i & 31);
    const float di = dinv[node];
    const float d2 = di * di;
    float4 hv = ((const float4*)h)[i];
    float4 av = ((const float4*)agg)[i];
    const float* bp = bias + c * 4;
    float4 o;
    o.x = fmaxf(av.x + d2 * hv.x + bp[0], 0.0f);
    o.y = fmaxf(av.y + d2 * hv.y + bp[1], 0.0f);
    o.z = fmaxf(av.z + d2 * hv.z + bp[2], 0.0f);
    o.w = fmaxf(av.w + d2 * hv.w + bp[3], 0.0f);
    ((float4*)h)[i] = o;
}

// ---------------------------------------------------------------------------
// Global mean pool: sum + counts via atomics, then divide.
// ---------------------------------------------------------------------------
__global__ __launch_bounds__(256) void pool_sum(const float* __restrict__ h,
                                                const int* __restrict__ batch,
                                                float* __restrict__ gsum,
                                                float* __restrict__ gcnt, int nNodes) {
    long i = (long)blockIdx.x * 256 + threadIdx.x;
    long total = (long)nNodes * 32;
    if (i >= total) return;
    const int node = (int)(i >> 5);
    const int c    = (int)(i & 31);
    const int b    = batch[node];
    float4 v = ((const float4*)h)[i];
    float* gp = gsum + (size_t)b * 128 + c * 4;
    atomicAdd(gp + 0, v.x);
    atomicAdd(gp + 1, v.y);
    atomicAdd(gp + 2, v.z);
    atomicAdd(gp + 3, v.w);
    if (c == 0) atomicAdd(&gcnt[b], 1.0f);
}

__global__ __launch_bounds__(256) void pool_div(const float* __restrict__ gsum,
                                                const float* __restrict__ gcnt,
                                                float* __restrict__ g, int G) {
    int i = blockIdx.x * 256 + threadIdx.x;
    if (i < G * 128) {
        int b = i >> 7;
        g[i] = gsum[i] / fmaxf(gcnt[b], 1.0f);
    }
}

// ---------------------------------------------------------------------------
// Launch
// Inputs: x, edge_index, batch, W1, b1, W2, b2, Wl1, bl1, Wl2, bl2
// ---------------------------------------------------------------------------
extern "C" void kernel_launch(void* const* d_in, const int* in_sizes, int n_in,
                              void* d_out, int out_size, void* d_ws, size_t ws_size,
                              hipStream_t stream) {
    const float* x     = (const float*)d_in[0];
    const int*   ei    = (const int*)d_in[1];
    const int*   batch = (const int*)d_in[2];
    const float* W1    = (const float*)d_in[3];
    const float* b1    = (const float*)d_in[4];
    const float* W2    = (const float*)d_in[5];
    const float* b2    = (const float*)d_in[6];
    const float* Wl1   = (const float*)d_in[7];
    const float* bl1   = (const float*)d_in[8];
    const float* Wl2   = (const float*)d_in[9];
    const float* bl2   = (const float*)d_in[10];
    float* out = (float*)d_out;

    const int nN = in_sizes[0] / 128;   // 100000
    const int nE = in_sizes[1] / 2;     // 1600000
    const int G  = out_size / 32;       // 256
    const int* src = ei;
    const int* dst = ei + nE;

    char* ws = (char*)d_ws;
    auto alloc = [&](size_t bytes) -> float* {
        float* p = (float*)ws;
        ws += (bytes + 255) & ~(size_t)255;
        return p;
    };
    float* bufA = alloc((size_t)nN * 128 * 4);  // h (gemm out), layer act
    float* bufB = alloc((size_t)nN * 128 * 4);  // agg
    float* bufC = alloc((size_t)nN * 128 * 4);  // layer-2 h / act
    float* deg  = alloc((size_t)nN * 4);
    float* dinv = alloc((size_t)nN * 4);
    float* gsum = alloc((size_t)G * 128 * 4);
    float* gcnt = alloc((size_t)G * 4);
    float* g    = alloc((size_t)G * 128 * 4);
    float* m1   = alloc((size_t)G * 128 * 4);

    const long nel = (long)nN * 128;
    const int  fB  = (int)((nel + 255) / 256);
    const int  eB  = (nE + 255) / 256;
    const int  nB  = (nN + 255) / 256;
    const int  qB  = (int)(((long)nN * 32 + 255) / 256);  // float4-granular passes
    const int  sB  = (nE + 7) / 8;                        // edge scatter blocks

    // Degree / normalization
    fill_f32<<<nB, 256, 0, stream>>>(deg, 1.0f, nN);
    fill_f32<<<fB, 256, 0, stream>>>(bufB, 0.0f, nel);
    fill_f32<<<(G * 128 + 255) / 256, 256, 0, stream>>>(gsum, 0.0f, (long)G * 128);
    fill_f32<<<(G + 255) / 256, 256, 0, stream>>>(gcnt, 0.0f, G);
    deg_count<<<eB, 256, 0, stream>>>(dst, deg, nE);
    rsqrt_k<<<nB, 256, 0, stream>>>(deg, dinv, nN);

    // Layer 1: h = x@W1; agg = scatter; act = relu(agg + dinv^2 h + b1)
    gemm_k128<<<(nN + 15) / 16, 256, 0, stream>>>(x, W1, nullptr, bufA, nN, 128, 0, 0);
    edge_scatter<<<sB, 256, 0, stream>>>(src, dst, dinv, bufA, bufB, nE);
    finalize_k<<<qB, 256, 0, stream>>>(bufB, bufA, dinv, b1, nN);

    // Layer 2
    fill_f32<<<fB, 256, 0, stream>>>(bufB, 0.0f, nel);
    gemm_k128<<<(nN + 15) / 16, 256, 0, stream>>>(bufA, W2, nullptr, bufC, nN, 128, 0, 0);
    edge_scatter<<<sB, 256, 0, stream>>>(src, dst, dinv, bufC, bufB, nE);
    finalize_k<<<qB, 256, 0, stream>>>(bufB, bufC, dinv, b2, nN);

    // Global mean pool
    pool_sum<<<qB, 256, 0, stream>>>(bufC, batch, gsum, gcnt, nN);
    pool_div<<<(G * 128 + 255) / 256, 256, 0, stream>>>(gsum, gcnt, g, G);

    // MLP head: relu(g@Wl1+bl1) @ Wl2 + bl2
    gemm_k128<<<(G + 15) / 16, 256, 0, stream>>>(g, Wl1, bl1, m1, G, 128, 1, 1);
    gemm_k128<<<(G + 15) / 16, 256, 0, stream>>>(m1, Wl2, bl2, out, G, 32, 0, 1);
}